// SparseScatter_73967926772532
// MI455X (gfx1250) — compile-verified
//
#include <hip/hip_runtime.h>
#include <stdint.h>

// Problem constants (from the reference setup).
#define NC    64          // channels
#define NH    512
#define NW    512
#define NBLK  4096        // number of scattered tiles
#define SLOTS 8192        // 8 * 32 * 32 output tile slots (exactly tile y)

typedef float v4f __attribute__((ext_vector_type(4)));  // native vec: NT-builtin OK

// One workgroup per 16x16x64 output tile slot.
// Fuses y_base->y copy with deterministic gather of all contributing x tiles.
// x tiles are staged via gfx1250 async global->LDS DMA (double buffered).
__global__ __launch_bounds__(256) void SparseScatter_gather_kernel(
    const float* __restrict__ x,       // [4096, 64, 16, 16]
    const float* __restrict__ y_base,  // [8, 64, 512, 512]
    const int*   __restrict__ idx,     // [4096, 3] (b, rblk, cblk)
    float* __restrict__ y_out)         // [8, 64, 512, 512]
{
  const int s = blockIdx.x;            // slot id
  const int t = threadIdx.x;           // 0..255
  const int b  = s >> 10;              // 1024 slots per batch image
  const int rb = (s >> 5) & 31;
  const int cb = s & 31;
  const int r0 = rb * 16;
  const int w0 = cb * 16;

  __shared__ int slist[NBLK];                    // contribution list (worst case)
  __shared__ int scnt[256];
  __shared__ int soff[256];
  __shared__ int s_total;
  __shared__ alignas(16) float xstage[2][4096];  // 2 x 16KB async staging buffers

  // ---- build contribution list: deterministic prefix-sum compaction ----
  // Thread t scans candidates [t*16, t*16+16) -> ascending n order overall.
  int local_n[16];
  int nl = 0;
  const int cbase = t * 16;
  #pragma unroll
  for (int i = 0; i < 16; ++i) {
    const int n  = cbase + i;
    const int bb = idx[n * 3 + 0];
    const int rr = idx[n * 3 + 1];
    const int cc = idx[n * 3 + 2];
    const int slot = ((bb << 5) + rr) * 32 + cc;
    if (slot == s) local_n[nl++] = n;
  }
  scnt[t] = nl;
  __syncthreads();
  if (t == 0) {
    int run = 0;
    for (int i = 0; i < 256; ++i) { soff[i] = run; run += scnt[i]; }
    s_total = run;
  }
  __syncthreads();
  {
    const int o = soff[t];
    for (int i = 0; i < nl; ++i) slist[o + i] = local_n[i];
  }
  __syncthreads();
  const int nC = s_total;

  // ---- load y_base tile into 64 accumulator VGPRs (16 v4f / thread) ----
  // q = k*256 + t indexes the 4096 float4s of this slot:
  //   c = q>>6, h = (q&63)>>2, j = q&3  (j = 16B segment within the 64B row)
  const v4f* y4 = reinterpret_cast<const v4f*>(y_base);
  v4f*       o4 = reinterpret_cast<v4f*>(y_out);
  v4f acc[16];
  int yidx[16];
  #pragma unroll
  for (int k = 0; k < 16; ++k) {
    const int q = k * 256 + t;
    const int c = q >> 6;
    const int rem = q & 63;
    const int h = rem >> 2;
    const int j = rem & 3;
    const int fi = ((b * NC + c) * NH + (r0 + h)) * (NW / 4) + (w0 >> 2) + j;
    yidx[k] = fi;
    acc[k] = __builtin_nontemporal_load(&y4[fi]);  // streaming: no reuse
  }

  // ---- accumulate contributions via async global->LDS staging ----
  // Each x tile = 64KB, staged in 4 chunks of 16KB (16 channels each).
  // Per chunk: 4 async b128 instructions per thread; each thread later reads
  // back exactly the LDS float4s it staged (wave-private -> only ASYNCcnt
  // sync needed, no cross-wave barrier in the pipeline).
  auto stage_chunk = [&](int e, int g, int buf) {
    const int n = slist[e];
    const char* gp = reinterpret_cast<const char*>(x)
                   + (size_t)n * 65536u + (size_t)g * 16384u;
    const uint32_t lb = (uint32_t)(uintptr_t)(&xstage[buf][0]);
    #pragma unroll
    for (int i = 0; i < 4; ++i) {
      const uint32_t la = lb + (uint32_t)((i * 256 + t) * 16);
      const uint64_t ga = (uint64_t)(uintptr_t)(gp + (i * 256 + t) * 16);
      asm volatile("global_load_async_to_lds_b128 %0, %1, off"
                   :: "v"(la), "v"(ga) : "memory");
    }
  };

  const int nchunks = nC * 4;
  if (nchunks > 0) {
    stage_chunk(0, 0, 0);                       // prologue: chunk 0 -> buf 0
    for (int jj = 0; jj < nchunks; ++jj) {
      const int buf = jj & 1;
      if (jj + 1 < nchunks) {
        const int nx = jj + 1;
        stage_chunk(nx >> 2, nx & 3, nx & 1);   // prefetch next chunk
        // in-order completion: <=4 outstanding => current chunk's 4 are done
        asm volatile("s_wait_asynccnt 0x4" ::: "memory");
      } else {
        asm volatile("s_wait_asynccnt 0x0" ::: "memory");
      }
      const int g = jj & 3;                     // channel group of this chunk
      const v4f* st = reinterpret_cast<const v4f*>(&xstage[buf][0]);
      #pragma unroll
      for (int i = 0; i < 4; ++i) {
        const v4f v = st[i * 256 + t];          // ds_load_b128
        const int k = g * 4 + i;
        acc[k] += v;
      }
    }
  }

  // ---- store result tile (streaming, non-temporal) ----
  #pragma unroll
  for (int k = 0; k < 16; ++k) {
    __builtin_nontemporal_store(acc[k], &o4[yidx[k]]);
  }
}

extern "C" void kernel_launch(void* const* d_in, const int* in_sizes, int n_in,
                              void* d_out, int out_size, void* d_ws, size_t ws_size,
                              hipStream_t stream) {
  const float* x      = (const float*)d_in[0];
  const float* y_base = (const float*)d_in[1];
  const int*   idx    = (const int*)d_in[2];
  (void)in_sizes; (void)n_in; (void)out_size; (void)d_ws; (void)ws_size;

  SparseScatter_gather_kernel<<<SLOTS, 256, 0, stream>>>(
      x, y_base, idx, (float*)d_out);
}